// RobotTrustModel_56710748176950
// MI455X (gfx1250) — compile-verified
//
#include <hip/hip_runtime.h>
#include <hip/hip_bf16.h>
#include <stdint.h>

// ---------------------------------------------------------------------------
// RobotTrustModel trust kernel for MI455X (gfx1250).
//
// out[r] = (f0(bc[i0]) * will) * f1(bc[i1])  -- factorized into two 100-entry
// LDS LUTs. The hot loop is HBM streaming: 64 MB of int32 index pairs in via
// CDNA5 async global->LDS (b128/lane, ASYNCcnt), 32 MB out via b128 stores.
// Per-wave 4-deep, 1KB-tile async ring; steady loop is branch-free with a
// fixed s_wait_asynccnt, followed by a short drain.
// Roofline: 96 MB @ 23.3 TB/s ~= 4.1 us; no matmul -> WMMA not applicable.
// ---------------------------------------------------------------------------

#define WAVES_PER_BLOCK 8      // 256 threads, wave32
#define DEPTH           4      // async ring depth (tiles in flight per wave)
#define TILE_PAIRS      128    // 32 lanes x 4 index-pairs per tile
#define TILE_BYTES      1024   // 128 pairs * 8 bytes
#define LOADS_PER_TILE  2      // two b128 async loads per lane per tile

__device__ __forceinline__ float sigmoid_f(float x) {
    return 1.0f / (1.0f + expf(-x));
}

// LDS byte offset of a generic pointer to a __shared__ object.
__device__ __forceinline__ uint32_t lds_off(const void* p) {
    return (uint32_t)(uint64_t)(const __attribute__((address_space(3))) char*)p;
}

__device__ __forceinline__ void async_load_b128(uint32_t lds_byte_off, uint64_t gaddr) {
    // GLOBAL_LOAD_ASYNC_TO_LDS_B128 (GV mode): vdst = per-lane LDS byte
    // offset, vaddr = 64-bit global address. Tracked by ASYNCcnt.
    asm volatile("global_load_async_to_lds_b128 %0, %1, off"
                 :
                 : "v"(lds_byte_off), "v"(gaddr)
                 : "memory");
}

__global__ __launch_bounds__(256) void robot_trust_kernel(
    const float* __restrict__ bin_centers,   // [n_bins]
    const int*   __restrict__ idxs,          // [n_pairs, 2] int32
    const float* __restrict__ pre_beta,      // [2]
    const float* __restrict__ pre_can_l,     // [2]
    const float* __restrict__ pre_can_u,     // [2]
    const float* __restrict__ pre_will_l,    // [1]
    const float* __restrict__ pre_will_u,    // [1]
    float*       __restrict__ out,           // [n_pairs]
    long long n_pairs, int n_bins)
{
    __shared__ float tab0[128];  // f0(bc[b]) * will
    __shared__ float tab1[128];  // f1(bc[b])
    __shared__ uint4 stage[WAVES_PER_BLOCK][DEPTH][64];  // 32 KB staging

    const int tid = threadIdx.x;

    // ---- Build the per-cap LUTs (threads 0..2*n_bins-1) -------------------
    if (tid < 2 * n_bins) {
        const int c = (tid < n_bins) ? 0 : 1;
        const int b = (tid < n_bins) ? tid : tid - n_bins;

        const float p   = bin_centers[b];
        const float pl  = pre_can_l[c];
        const float pu  = pre_can_u[c];
        const float cl  = sigmoid_f(fminf(pl, pu));
        const float cu  = sigmoid_f(fmaxf(pl, pu));
        const float bta = pre_beta[c] * pre_beta[c];
        const float den = cu - cl;

        // smooth branch kept for fidelity (dead for beta = pre_beta^2 >= 0)
        const float smooth = 1.0f - (1.0f / (bta * den)) *
            (log1pf(expf(bta * (p - cl))) - log1pf(expf(bta * (p - cu))));
        const float linear = (cu - p) / (den + 1e-4f);
        float f = (bta < -50.0f)
                      ? smooth
                      : ((p <= cl) ? 1.0f : ((p > cu) ? 0.0f : linear));

        if (c == 0) {
            const float wl   = pre_will_l[0];
            const float wu   = pre_will_u[0];
            const float will = 0.5f * (sigmoid_f(fminf(wl, wu)) +
                                       sigmoid_f(fmaxf(wl, wu)));
            tab0[b] = f * will;
        } else {
            tab1[b] = f;
        }
    }
    __syncthreads();

    // ---- Async-pipelined streaming over full 128-pair tiles ---------------
    const int       lane   = tid & 31;
    const int       wave   = tid >> 5;
    const long long gwave  = (long long)blockIdx.x * WAVES_PER_BLOCK + wave;
    const long long nwaves = (long long)gridDim.x * WAVES_PER_BLOCK;
    const long long ntiles = n_pairs / TILE_PAIRS;

    const long long mytiles =
        (gwave < ntiles) ? ((ntiles - 1 - gwave) / nwaves + 1) : 0;

    // Lane owns bytes [32*lane, 32*lane+32) of each tile buffer.
    const uint32_t lane_lds = lds_off(&stage[wave][0][0]) + (uint32_t)lane * 32u;
    const uint64_t gbase    = (uint64_t)idxs;

    // Prologue: fill the ring (2 async b128 loads per tile per lane).
    const long long pre = (mytiles < DEPTH) ? mytiles : DEPTH;
    for (long long k = 0; k < pre; ++k) {
        const uint64_t ga = gbase +
            (uint64_t)(gwave + k * nwaves) * TILE_BYTES + (uint64_t)lane * 32u;
        const uint32_t lo = lane_lds + (uint32_t)k * TILE_BYTES;
        async_load_b128(lo,      ga);
        async_load_b128(lo + 16, ga + 16);
    }

    const uint64_t  g_istep = (uint64_t)nwaves * TILE_BYTES;
    const long long o_step  = nwaves * TILE_PAIRS;
    uint64_t g_issue = gbase +
        (uint64_t)(gwave + (long long)DEPTH * nwaves) * TILE_BYTES +
        (uint64_t)lane * 32u;
    float* optr = out + gwave * TILE_PAIRS + lane * 4;

    int buf = 0;

    // Steady state: inflight == DEPTH tiles (2*DEPTH async ops); the oldest
    // tile is complete once asynccnt <= 2*DEPTH - 2 == 6. Branch-free wait.
    const long long steady = (mytiles > DEPTH) ? (mytiles - DEPTH) : 0;
    for (long long i = 0; i < steady; ++i) {
        asm volatile("s_wait_asynccnt 6" ::: "memory");

        const uint4 a = stage[wave][buf][2 * lane + 0];
        const uint4 b = stage[wave][buf][2 * lane + 1];
        float4 o;
        o.x = tab0[a.x] * tab1[a.y];
        o.y = tab0[a.z] * tab1[a.w];
        o.z = tab0[b.x] * tab1[b.y];
        o.w = tab0[b.z] * tab1[b.w];
        *(float4*)optr = o;

        // Refill this buffer with tile i+DEPTH.
        const uint32_t lo = lane_lds + (uint32_t)buf * TILE_BYTES;
        async_load_b128(lo,      g_issue);
        async_load_b128(lo + 16, g_issue + 16);

        g_issue += g_istep;
        optr    += o_step;
        buf      = (buf + 1) & (DEPTH - 1);
    }

    // Drain: everything outstanding has been issued; wait once, consume rest.
    asm volatile("s_wait_asynccnt 0" ::: "memory");
    for (long long i = steady; i < mytiles; ++i) {
        const uint4 a = stage[wave][buf][2 * lane + 0];
        const uint4 b = stage[wave][buf][2 * lane + 1];
        float4 o;
        o.x = tab0[a.x] * tab1[a.y];
        o.y = tab0[a.z] * tab1[a.w];
        o.z = tab0[b.x] * tab1[b.y];
        o.w = tab0[b.z] * tab1[b.w];
        *(float4*)optr = o;

        optr += o_step;
        buf   = (buf + 1) & (DEPTH - 1);
    }

    // ---- Remainder pairs (n_pairs % 128), direct path ---------------------
    const long long rem_start = ntiles * TILE_PAIRS;
    for (long long p = rem_start + (long long)blockIdx.x * blockDim.x + tid;
         p < n_pairs;
         p += (long long)gridDim.x * blockDim.x) {
        const int i0 = idxs[2 * p + 0];
        const int i1 = idxs[2 * p + 1];
        out[p] = tab0[i0] * tab1[i1];
    }
}

extern "C" void kernel_launch(void* const* d_in, const int* in_sizes, int n_in,
                              void* d_out, int out_size, void* d_ws, size_t ws_size,
                              hipStream_t stream) {
    (void)n_in; (void)d_ws; (void)ws_size;

    const float* bin_centers = (const float*)d_in[0];
    const int*   idxs        = (const int*)d_in[1];
    const float* pre_beta    = (const float*)d_in[2];
    const float* pre_can_l   = (const float*)d_in[3];
    const float* pre_can_u   = (const float*)d_in[4];
    const float* pre_will_l  = (const float*)d_in[5];
    const float* pre_will_u  = (const float*)d_in[6];
    float*       out         = (float*)d_out;

    const int       n_bins  = in_sizes[0];
    const long long n_pairs = (long long)in_sizes[1] / 2;  // == out_size

    // Target ~16 tiles per wave for a deep, steady async pipeline.
    const long long ntiles = n_pairs / TILE_PAIRS;
    int blocks = (int)((ntiles + (long long)WAVES_PER_BLOCK * 16 - 1) /
                       ((long long)WAVES_PER_BLOCK * 16));
    if (blocks < 1)    blocks = 1;
    if (blocks > 4096) blocks = 4096;

    robot_trust_kernel<<<blocks, 256, 0, stream>>>(
        bin_centers, idxs, pre_beta, pre_can_l, pre_can_u,
        pre_will_l, pre_will_u, out, n_pairs, n_bins);
    (void)out_size;
}